// OpenLSTM_18322330485250
// MI455X (gfx1250) — compile-verified
//
#include <hip/hip_runtime.h>

// CDNA5 / gfx1250 projected-LSTM. One wave32 owns 16 sequences.
//
// Per step: gates(64x16batch) = 4x V_WMMA_F32_16X16X4_F32 with
//   A = constant weight tile (16 gates x 4 features, lives in VGPRs),
//   B = per-step inputs (4 x 16 batch): lanes 0-15 carry K=0,1 (x0,x1),
//       lanes 16-31 carry K=2,3 (h0,h1  -- y_t during teacher forcing).
// C/D layout => lane = batch, VGPR j = gate unit (j + 8*half): the cell
// update is pure elementwise (native v_tanh_f32) with 8-way ILP, the
// output projection is 8 register FMAs + ONE shfl_xor(16) per component,
// and the h feedback lives in the upper half-wave's registers.
// Input loads are software-pipelined one step ahead so global latency
// hides under the previous step's VALU work; no LDS, no barriers.

typedef __attribute__((ext_vector_type(2))) float v2f;
typedef __attribute__((ext_vector_type(8))) float v8f;

#define T_LEN 1024
#define N_CTX 256

#if __has_builtin(__builtin_amdgcn_tanhf)
__device__ __forceinline__ float tanh_fast(float x) { return __builtin_amdgcn_tanhf(x); }
#define HAVE_NATIVE_TANH 1
#elif __has_builtin(__builtin_amdgcn_tanh_f32)
__device__ __forceinline__ float tanh_fast(float x) { return __builtin_amdgcn_tanh_f32(x); }
#define HAVE_NATIVE_TANH 1
#else
__device__ __forceinline__ float tanh_fast(float x) {
  float e = __builtin_amdgcn_exp2f(-2.88539008177792681472f * x);
  return __fmaf_rn(2.0f, __builtin_amdgcn_rcpf(1.0f + e), -1.0f);
}
#define HAVE_NATIVE_TANH 0
#endif

__device__ __forceinline__ float sigmoid_fast(float x) {
#if HAVE_NATIVE_TANH
  return __fmaf_rn(0.5f, tanh_fast(0.5f * x), 0.5f);  // v_tanh_f32 path
#else
  float e = __builtin_amdgcn_exp2f(-1.44269504088896340736f * x);
  return __builtin_amdgcn_rcpf(1.0f + e);
#endif
}

__global__ __launch_bounds__(32, 1) void OpenLSTM_cdna5_kernel(
    const float* __restrict__ u,      // (B, 1024, 4): [x0,x1,y0,y1]
    const float* __restrict__ w_ih,   // (64, 2)
    const float* __restrict__ w_hh,   // (64, 2)
    const float* __restrict__ b_ih,   // (64)
    const float* __restrict__ b_hh,   // (64)
    const float* __restrict__ w_hr,   // (2, 16)
    float* __restrict__ out)          // (B, 1024, 2)
{
  const int lane = threadIdx.x & 31;
  const int half = lane >> 4;         // which K-pair / which unit group
  const int l16  = lane & 15;         // this lane's batch within the tile
  const long b0  = (long)blockIdx.x * 16;

  // ---- constant A-operand tiles (16x4 f32, 2 VGPRs each) ----
  // A_n[M][K] = W[g=16n+M][K], K in {w_ih0, w_ih1, w_hh0, w_hh1}.
  // 16x4 A layout: lanes 0-15 hold M=lane, K=0 (.x) / K=1 (.y);
  //                lanes 16-31 hold M=lane-16, K=2 (.x) / K=3 (.y).
  v2f wa[4];
#pragma unroll
  for (int n = 0; n < 4; ++n) {
    const int g = 16 * n + l16;
    const float* src = half ? w_hh : w_ih;
    wa[n].x = src[2 * g + 0];
    wa[n].y = src[2 * g + 1];
  }

  // C/D layout: lane = batch N=l16, VGPR j = unit m = j + 8*half.
  float bia[4][8];                    // bias per (gate type, unit j)
  float w0[8], w1[8];                 // w_hr rows for this lane's units
#pragma unroll
  for (int j = 0; j < 8; ++j) {
    const int m = j + 8 * half;
#pragma unroll
    for (int n = 0; n < 4; ++n) bia[n][j] = b_ih[16 * n + m] + b_hh[16 * n + m];
    w0[j] = w_hr[m];
    w1[j] = w_hr[16 + m];
  }

  float c[8];                         // cell state: batch l16, unit j+8*half
#pragma unroll
  for (int j = 0; j < 8; ++j) c[j] = 0.0f;

  // lanes 0-15 stream {x0,x1}; lanes 16-31 stream {y0,y1} (+8 bytes)
  const float* uptr = u + (long)(b0 + l16) * (T_LEN * 4) + half * 2;
  float* optr = out + (long)(b0 + l16) * T_LEN * 2;

  float h0r = 0.0f, h1r = 0.0f;       // every lane ends up with its batch's h

  auto step = [&](int t, float2 xv, bool feedback) {
    v2f bop;                                      // B operand (4 x 16 batch)
    bop.x = (feedback && half) ? h0r : xv.x;      // v_cndmask, branchless
    bop.y = (feedback && half) ? h1r : xv.y;

    v8f ai, af, ag, ao;
#pragma unroll
    for (int j = 0; j < 8; ++j) {
      ai[j] = bia[0][j]; af[j] = bia[1][j]; ag[j] = bia[2][j]; ao[j] = bia[3][j];
    }
    ai = __builtin_amdgcn_wmma_f32_16x16x4_f32(false, wa[0], false, bop, (short)0, ai, false, false);
    af = __builtin_amdgcn_wmma_f32_16x16x4_f32(false, wa[1], false, bop, (short)0, af, false, false);
    ag = __builtin_amdgcn_wmma_f32_16x16x4_f32(false, wa[2], false, bop, (short)0, ag, false, false);
    ao = __builtin_amdgcn_wmma_f32_16x16x4_f32(false, wa[3], false, bop, (short)0, ao, false, false);

    // elementwise cell update + in-register projection dot (2-way FMA trees)
    float p0a = 0.f, p0b = 0.f, p1a = 0.f, p1b = 0.f;
#pragma unroll
    for (int j = 0; j < 8; ++j) {
      const float ig = sigmoid_fast(ai[j]);
      const float fg = sigmoid_fast(af[j]);
      const float zg = tanh_fast(ag[j]);
      const float og = sigmoid_fast(ao[j]);
      c[j] = __fmaf_rn(fg, c[j], ig * zg);
      const float s = og * tanh_fast(c[j]);
      if (j & 1) { p0b = __fmaf_rn(s, w0[j], p0b); p1b = __fmaf_rn(s, w1[j], p1b); }
      else       { p0a = __fmaf_rn(s, w0[j], p0a); p1a = __fmaf_rn(s, w1[j], p1a); }
    }
    const float p0 = p0a + p0b;
    const float p1 = p1a + p1b;
    // combine the two 8-unit half-sums: ONE cross-half exchange per component
    h0r = p0 + __shfl_xor(p0, 16, 32);
    h1r = p1 + __shfl_xor(p1, 16, 32);

    if (!half)                                    // 8B store per sequence
      *(float2*)(optr + 2 * t) = make_float2(h0r, h1r);
  };

  // ---- software pipeline: input for step t+1 loads under step t's VALU ----
  float2 xv_cur = *(const float2*)(uptr);

  // phase 1: teacher forcing (h := y_t from data; steps chain only through c,
  // so a 2x unroll lets WMMAs of t+1 co-issue with transcendentals of t)
#pragma unroll 2
  for (int t = 0; t < N_CTX; ++t) {
    float2 xv_nxt = *(const float2*)(uptr + 4 * (t + 1));  // t+1 <= 256: in range
    __builtin_prefetch(uptr + 4 * (t + 24), 0, 1);         // global_prefetch_b8
    step(t, xv_cur, false);
    xv_cur = xv_nxt;
  }

  // phase 2: free running (h fed back from the upper half-wave's registers);
  // last iteration peeled so the lookahead load never runs past the buffer
  for (int t = N_CTX; t < T_LEN - 1; ++t) {
    float2 xv_nxt = *(const float2*)(uptr + 4 * (t + 1));
    if (t + 24 < T_LEN) __builtin_prefetch(uptr + 4 * (t + 24), 0, 1);
    step(t, xv_cur, true);
    xv_cur = xv_nxt;
  }
  step(T_LEN - 1, xv_cur, true);
}

extern "C" void kernel_launch(void* const* d_in, const int* in_sizes, int n_in,
                              void* d_out, int out_size, void* d_ws, size_t ws_size,
                              hipStream_t stream) {
  const float* u    = (const float*)d_in[0];
  const float* w_ih = (const float*)d_in[1];
  const float* w_hh = (const float*)d_in[2];
  const float* b_ih = (const float*)d_in[3];
  const float* b_hh = (const float*)d_in[4];
  const float* w_hr = (const float*)d_in[5];
  float* out = (float*)d_out;

  const int B = in_sizes[0] / (T_LEN * 4);   // 4096
  dim3 grid(B / 16), block(32);
  hipLaunchKernelGGL(OpenLSTM_cdna5_kernel, grid, block, 0, stream,
                     u, w_ih, w_hh, b_ih, b_hh, w_hr, out);
}